// RingEncoder_51393578664320
// MI455X (gfx1250) — compile-verified
//
#include <hip/hip_runtime.h>
#include <hip/hip_bf16.h>
#include <math.h>

// ---------------------------------------------------------------------------
// RingEncoder (PointNet-style) for MI455X / gfx1250, fp32 end-to-end using
// V_WMMA_F32_16X16X4_F32 for every K>=64 GEMM. Heavy kernels stage activation
// tiles into LDS with GLOBAL_LOAD_ASYNC_TO_LDS_B128 (ASYNCcnt path) and hoist
// weight fragments into registers. B=8, C=5, N=4096, R=16.
// ---------------------------------------------------------------------------

#define NPTS   4096
#define NB     16            // NPTS / 256 (light kernels' n-block)
#define NBLK64 64            // NPTS / 64  (staged kernels' n-block)
#define BATCH  8
#define EPS    1e-5f
#define OUT_CH 1216
#define LDP    72            // 64-col tile + 8 pad: rows k,k+2 differ by 16 banks
#define OUT_TRANS_OFF ((size_t)BATCH * OUT_CH * NPTS)        // 39845888
#define OUT_TF_OFF    (OUT_TRANS_OFF + (size_t)BATCH * 9)    // +72

typedef __attribute__((ext_vector_type(2))) float v2f;
typedef __attribute__((ext_vector_type(8))) float v8f;
typedef __attribute__((ext_vector_type(4))) int   v4i;

#define AS1 __attribute__((address_space(1)))
#define AS3 __attribute__((address_space(3)))

#if defined(__gfx1250__) && __has_builtin(__builtin_amdgcn_global_load_async_to_lds_b128)
#define USE_ASYNC_LDS 1
#else
#define USE_ASYNC_LDS 0
#endif

// D = A(16x4) * B(4x16) + C(16x16), fp32, wave32.
__device__ __forceinline__ v8f wmma4(v2f a, v2f b, v8f c) {
  return __builtin_amdgcn_wmma_f32_16x16x4_f32(false, a, false, b, (short)0, c,
                                               false, false);
}

// 16B global -> LDS copy, async (ASYNCcnt) when available.
__device__ __forceinline__ void copy16_to_lds(const float* g, float* l) {
#if USE_ASYNC_LDS
  __builtin_amdgcn_global_load_async_to_lds_b128((AS1 v4i*)g, (AS3 v4i*)l, 0, 0);
#else
  *(float4*)l = *(const float4*)g;
#endif
}

__device__ __forceinline__ void lds_stage_fence() {
#if USE_ASYNC_LDS
#if __has_builtin(__builtin_amdgcn_s_wait_asynccnt)
  __builtin_amdgcn_s_wait_asynccnt(0);
#else
  asm volatile("s_wait_asynccnt 0x0" ::: "memory");
#endif
#endif
  __syncthreads();
}

// ---------------- Cin=5 conv + BN + ReLU (VALU; K too small for WMMA) -------
__global__ void conv5_bn_relu_kernel(const float* __restrict__ in,   // [B][5][N]
                                     const float* __restrict__ W,    // [O][5]
                                     const float* __restrict__ bias,
                                     const float* __restrict__ g,
                                     const float* __restrict__ bb,
                                     const float* __restrict__ bm,
                                     const float* __restrict__ bv,
                                     float* __restrict__ out, int O) {
  int n = blockIdx.x * 256 + threadIdx.x;
  int o = blockIdx.y;
  int b = blockIdx.z;
  const float* ip = in + (size_t)b * 5 * NPTS + n;
  float acc = 0.f;
#pragma unroll
  for (int c = 0; c < 5; ++c) acc = fmaf(W[o * 5 + c], ip[(size_t)c * NPTS], acc);
  float s = g[o] / sqrtf(bv[o] + EPS);
  float val = (acc + bias[o] - bm[o]) * s + bb[o];
  out[((size_t)b * O + o) * NPTS + n] = fmaxf(val, 0.f);
}

// ---------------- generic per-point GEMM + BN (+ReLU), WMMA fp32 -----------
// out[b][m][n] = act(bn(sum_k W[m][k] * in[b][k][n] + bias[m]))
// grid = (N/16, O/16, B), block = 32 (one wave per 16x16 tile).
__global__ void convbn_wmma_kernel(const float* __restrict__ in, size_t inBStride,
                                   const float* __restrict__ W,
                                   const float* __restrict__ bias,
                                   const float* __restrict__ g,
                                   const float* __restrict__ bb,
                                   const float* __restrict__ bm,
                                   const float* __restrict__ bv,
                                   float* __restrict__ out, int K, int O,
                                   int relu) {
  const int lane = threadIdx.x & 31;
  const int half = lane >> 4;
  const int r    = lane & 15;
  const int n0 = blockIdx.x * 16;
  const int m0 = blockIdx.y * 16;
  const int b  = blockIdx.z;
  const float* inb = in + (size_t)b * inBStride + n0 + r;
  const float* wr  = W + (size_t)(m0 + r) * K;
  v8f acc = {};
  for (int k = 0; k < K; k += 4) {
    int ka = k + 2 * half;
    v2f a, bf;
    a.x  = wr[ka];
    a.y  = wr[ka + 1];
    bf.x = inb[(size_t)ka * NPTS];
    bf.y = inb[(size_t)(ka + 1) * NPTS];
    __builtin_prefetch(inb + (size_t)(ka + 4) * NPTS, 0, 1);  // global_prefetch_b8
    acc = wmma4(a, bf, acc);
  }
#pragma unroll
  for (int v = 0; v < 8; ++v) {
    int m = m0 + v + 8 * half;
    float s = g[m] / sqrtf(bv[m] + EPS);
    float val = (acc[v] + bias[m] - bm[m]) * s + bb[m];
    if (relu) val = fmaxf(val, 0.f);
    out[((size_t)b * O + m) * NPTS + n0 + r] = val;
  }
}

// ------- GEMM (K=128) + BN (+ReLU) fused with max-over-N -> partials --------
// LDS-staged: block = 8 waves = 128 output rows x 64 points; the 128x64
// activation tile is async-copied to LDS once and shared by all waves; each
// wave's 16x128 weight panel is hoisted into 32 v2f register fragments.
// grid = (NBLK64, O/128, B), block = 256.
__global__ void convbn_max_wmma_kernel(const float* __restrict__ in,   // [B][128][N]
                                       const float* __restrict__ W,    // [O][128]
                                       const float* __restrict__ bias,
                                       const float* __restrict__ g,
                                       const float* __restrict__ bb,
                                       const float* __restrict__ bm,
                                       const float* __restrict__ bv,
                                       float* __restrict__ partial,    // [B][O][NBLK64]
                                       int O, int relu) {
  __shared__ float tile[128 * LDP];   // 36 KB
  const int tid  = threadIdx.x;
  const int wave = tid >> 5;
  const int lane = tid & 31;
  const int half = lane >> 4;
  const int r    = lane & 15;
  const int nblk = blockIdx.x;
  const int b    = blockIdx.z;
  const int m0   = (blockIdx.y * 8 + wave) * 16;
  const int n0   = nblk * 64;
  const int K    = 128;

  // ---- stage activation tile K(128) x 64 into LDS (async path) ----
  const float* gsrc = in + (size_t)b * K * NPTS + n0;
#pragma unroll
  for (int it = 0; it < 8; ++it) {
    int fl = (it * 256 + tid) * 4;        // float index within 128x64 tile
    int k  = fl >> 6;
    int c  = fl & 63;
    copy16_to_lds(gsrc + (size_t)k * NPTS + c, &tile[k * LDP + c]);
  }
  lds_stage_fence();

  // ---- hoist this wave's weight panel into registers ----
  const float* wr = W + (size_t)(m0 + r) * K;
  v2f aw[32];
#pragma unroll
  for (int kk = 0; kk < 32; ++kk) {
    int ka = kk * 4 + 2 * half;
    aw[kk].x = wr[ka];
    aw[kk].y = wr[ka + 1];
  }
  float sArr[8], tArr[8];
#pragma unroll
  for (int v = 0; v < 8; ++v) {
    int m = m0 + v + 8 * half;
    float s = g[m] / sqrtf(bv[m] + EPS);
    sArr[v] = s;
    tArr[v] = (bias[m] - bm[m]) * s + bb[m];
  }
  float vmax[8];
#pragma unroll
  for (int v = 0; v < 8; ++v) vmax[v] = -__builtin_inff();

  for (int st = 0; st < 4; ++st) {        // 4 n-subtiles of 16
    const float* lp = &tile[st * 16 + r];
    v8f acc = {};
#pragma unroll
    for (int kk = 0; kk < 32; ++kk) {
      int ka = kk * 4 + 2 * half;
      v2f bf;
      bf.x = lp[ka * LDP];
      bf.y = lp[(ka + 1) * LDP];
      acc = wmma4(aw[kk], bf, acc);
    }
#pragma unroll
    for (int v = 0; v < 8; ++v) {
      float val = acc[v] * sArr[v] + tArr[v];
      if (relu) val = fmaxf(val, 0.f);
      vmax[v] = fmaxf(vmax[v], val);
    }
  }
  // reduce over the 16 lanes of each half (each 16-group shares its M rows)
#pragma unroll
  for (int off = 8; off >= 1; off >>= 1)
#pragma unroll
    for (int v = 0; v < 8; ++v)
      vmax[v] = fmaxf(vmax[v], __shfl_xor(vmax[v], off));
  if (r == 0) {
#pragma unroll
    for (int v = 0; v < 8; ++v)
      partial[((size_t)b * O + m0 + v + 8 * half) * NBLK64 + nblk] = vmax[v];
  }
}

__global__ void reduce_max_kernel(const float* __restrict__ partial,
                                  float* __restrict__ out, int total, int nb) {
  int idx = blockIdx.x * 256 + threadIdx.x;
  if (idx >= total) return;
  const float* p = partial + (size_t)idx * nb;
  float m = p[0];
  for (int j = 1; j < nb; ++j) m = fmaxf(m, p[j]);
  out[idx] = m;
}

// ---------------- small FC + BN + ReLU (B=8, VALU) --------------------------
__global__ void fcbn_kernel(const float* __restrict__ in, const float* __restrict__ W,
                            const float* __restrict__ bias, const float* __restrict__ g,
                            const float* __restrict__ bb, const float* __restrict__ bm,
                            const float* __restrict__ bv, float* __restrict__ out,
                            int I, int O) {
  int idx = blockIdx.x * 256 + threadIdx.x;
  if (idx >= BATCH * O) return;
  int b = idx / O, o = idx % O;
  const float* ip = in + (size_t)b * I;
  const float* wp = W + (size_t)o * I;
  float acc = 0.f;
  for (int i = 0; i < I; ++i) acc = fmaf(ip[i], wp[i], acc);
  float s = g[o] / sqrtf(bv[o] + EPS);
  out[idx] = fmaxf((acc + bias[o] - bm[o]) * s + bb[o], 0.f);
}

__global__ void fceye_kernel(const float* __restrict__ in, const float* __restrict__ W,
                             const float* __restrict__ bias, float* __restrict__ out,
                             int I, int O, int k) {
  int idx = blockIdx.x * 256 + threadIdx.x;
  if (idx >= BATCH * O) return;
  int b = idx / O, o = idx % O;
  const float* ip = in + (size_t)b * I;
  const float* wp = W + (size_t)o * I;
  float acc = 0.f;
  for (int i = 0; i < I; ++i) acc = fmaf(ip[i], wp[i], acc);
  acc += bias[o];
  if ((o / k) == (o % k)) acc += 1.f;   // + eye(k)
  out[idx] = acc;
}

// ---------------- apply 3x3 input transform ---------------------------------
__global__ void apply_trans_kernel(const float* __restrict__ x,
                                   const float* __restrict__ trans,   // [B][3][3]
                                   float* __restrict__ xt) {          // [B][5][N]
  int n = blockIdx.x * 256 + threadIdx.x;
  int b = blockIdx.y;
  const float* t  = trans + b * 9;
  const float* xb = x + (size_t)b * 5 * NPTS;
  float* ob = xt + (size_t)b * 5 * NPTS;
  float x0 = xb[n], x1 = xb[NPTS + n], x2 = xb[2 * NPTS + n];
#pragma unroll
  for (int e = 0; e < 3; ++e)
    ob[(size_t)e * NPTS + n] = x0 * t[e] + x1 * t[3 + e] + x2 * t[6 + e];
  ob[(size_t)3 * NPTS + n] = xb[(size_t)3 * NPTS + n];
  ob[(size_t)4 * NPTS + n] = xb[(size_t)4 * NPTS + n];
}

// ------ feature transform: h'[b,e,n] = sum_c h[b,c,n]*TF[b,c,e] (WMMA) ------
// A[m,k] = TF[b][k][m] (transposed load). Writes straight into output rows 1152+.
__global__ void feattrans_wmma_kernel(const float* __restrict__ h,   // [B][64][N]
                                      const float* __restrict__ tf,  // [B][64][64]
                                      float* __restrict__ outMain) {
  const int lane = threadIdx.x & 31;
  const int half = lane >> 4;
  const int r    = lane & 15;
  const int n0 = blockIdx.x * 16;
  const int m0 = blockIdx.y * 16;
  const int b  = blockIdx.z;
  const float* tfb = tf + (size_t)b * 64 * 64;
  const float* hb  = h + (size_t)b * 64 * NPTS + n0 + r;
  v8f acc = {};
  for (int k = 0; k < 64; k += 4) {
    int ka = k + 2 * half;
    v2f a, bf;
    a.x  = tfb[(size_t)ka * 64 + m0 + r];
    a.y  = tfb[(size_t)(ka + 1) * 64 + m0 + r];
    bf.x = hb[(size_t)ka * NPTS];
    bf.y = hb[(size_t)(ka + 1) * NPTS];
    acc = wmma4(a, bf, acc);
  }
#pragma unroll
  for (int v = 0; v < 8; ++v) {
    int m = m0 + v + 8 * half;
    outMain[(size_t)b * OUT_CH * NPTS + (size_t)(1152 + m) * NPTS + n0 + r] = acc[v];
  }
}

// ------ ring projection (WMMA, LDS-staged) + masked max ---------------------
// BN scale is strictly positive, so max commutes with BN/bias: compute raw
// W_r . h', masked-max here, apply bias+BN on the pooled 16x128 later.
// grid = (NBLK64, 16, B), block = 256 (8 waves cover the 128 output rows).
__global__ void ring_wmma_kernel(const float* __restrict__ outMain,
                                 const float* __restrict__ rW,    // [16][128][64]
                                 const int* __restrict__ ring,    // [B][N]
                                 float* __restrict__ partial) {   // [B][16][128][NBLK64]
  __shared__ float tile[64 * LDP];   // 18 KB: h' tile 64 x 64
  const int tid  = threadIdx.x;
  const int wave = tid >> 5;
  const int lane = tid & 31;
  const int half = lane >> 4;
  const int r    = lane & 15;
  const int nblk = blockIdx.x;
  const int rr   = blockIdx.y;
  const int b    = blockIdx.z;
  const int m0   = wave * 16;
  const int n0   = nblk * 64;

  const float* hb = outMain + (size_t)b * OUT_CH * NPTS + (size_t)1152 * NPTS + n0;
#pragma unroll
  for (int it = 0; it < 4; ++it) {
    int fl = (it * 256 + tid) * 4;        // float index within 64x64 tile
    int k  = fl >> 6;
    int c  = fl & 63;
    copy16_to_lds(hb + (size_t)k * NPTS + c, &tile[k * LDP + c]);
  }
  lds_stage_fence();

  const float* wr = rW + ((size_t)rr * 128 + m0 + r) * 64;
  v2f aw[16];
#pragma unroll
  for (int kk = 0; kk < 16; ++kk) {
    int ka = kk * 4 + 2 * half;
    aw[kk].x = wr[ka];
    aw[kk].y = wr[ka + 1];
  }
  float vmax[8];
#pragma unroll
  for (int v = 0; v < 8; ++v) vmax[v] = -__builtin_inff();

  for (int st = 0; st < 4; ++st) {
    const float* lp = &tile[st * 16 + r];
    v8f acc = {};
#pragma unroll
    for (int kk = 0; kk < 16; ++kk) {
      int ka = kk * 4 + 2 * half;
      v2f bf;
      bf.x = lp[ka * LDP];
      bf.y = lp[(ka + 1) * LDP];
      acc = wmma4(aw[kk], bf, acc);
    }
    const int  rp   = ring[(size_t)b * NPTS + n0 + st * 16 + r];  // lane's column
    const bool keep = (rp == rr);
#pragma unroll
    for (int v = 0; v < 8; ++v)
      vmax[v] = fmaxf(vmax[v], keep ? acc[v] : -__builtin_inff());
  }
#pragma unroll
  for (int off = 8; off >= 1; off >>= 1)
#pragma unroll
    for (int v = 0; v < 8; ++v)
      vmax[v] = fmaxf(vmax[v], __shfl_xor(vmax[v], off));
  if (r == 0) {
#pragma unroll
    for (int v = 0; v < 8; ++v)
      partial[(((size_t)b * 16 + rr) * 128 + m0 + v + 8 * half) * NBLK64 + nblk] = vmax[v];
  }
}

__global__ void ring_reduce_bn_kernel(const float* __restrict__ partial,
                                      const float* __restrict__ rbias,  // [16][128]
                                      const float* __restrict__ g,
                                      const float* __restrict__ bb,
                                      const float* __restrict__ bm,
                                      const float* __restrict__ bv,
                                      float* __restrict__ pooled) {     // [B][16][128]
  int idx = blockIdx.x * 256 + threadIdx.x;
  if (idx >= BATCH * 16 * 128) return;
  int rd = idx % (16 * 128);
  const float* p = partial + (size_t)idx * NBLK64;
  float m = p[0];
  for (int j = 1; j < NBLK64; ++j) m = fmaxf(m, p[j]);
  float s = g[rd] / sqrtf(bv[rd] + EPS);
  pooled[idx] = (m + rbias[rd] - bm[rd]) * s + bb[rd];
}

__global__ void ring_scatter_kernel(const float* __restrict__ pooled,
                                    const int* __restrict__ ring,
                                    float* __restrict__ outMain) {
  int n = blockIdx.x * 256 + threadIdx.x;
  int d = blockIdx.y;
  int b = blockIdx.z;
  int rp = ring[(size_t)b * NPTS + n];
  outMain[(size_t)b * OUT_CH * NPTS + (size_t)(1024 + d) * NPTS + n] =
      pooled[((size_t)b * 16 + rp) * 128 + d];
}

__global__ void broadcast_kernel(const float* __restrict__ gvec,
                                 float* __restrict__ outMain) {
  int n = blockIdx.x * 256 + threadIdx.x;
  int o = blockIdx.y;
  int b = blockIdx.z;
  outMain[(size_t)b * OUT_CH * NPTS + (size_t)o * NPTS + n] = gvec[b * 1024 + o];
}

// ---------------------------------------------------------------------------
extern "C" void kernel_launch(void* const* d_in, const int* in_sizes, int n_in,
                              void* d_out, int out_size, void* d_ws, size_t ws_size,
                              hipStream_t stream) {
  (void)in_sizes; (void)n_in; (void)out_size; (void)ws_size;
  // Flattened input order (depth-first insertion order of setup_inputs):
  // 0:x 1:ring | stn:2..33 | conv1:34,35 bn1:36..39 | fstn:40..71 |
  // gpool:72..83 | ring.W:84 ring.b:85 ring.bn:86..89
  const float* x    = (const float*)d_in[0];
  const int*   ring = (const int*)d_in[1];
  auto F = [&](int i) { return (const float*)d_in[i]; };

  float* out   = (float*)d_out;
  float* trans = out + OUT_TRANS_OFF;   // (8,3,3)
  float* tf    = out + OUT_TF_OFF;      // (8,64,64)

  float* ws = (float*)d_ws;
  float* bufA    = ws; ws += (size_t)BATCH * 64 * NPTS;      // 8 MB
  float* bufB    = ws; ws += (size_t)BATCH * 128 * NPTS;     // 16 MB
  float* bufH    = ws; ws += (size_t)BATCH * 64 * NPTS;      // 8 MB
  float* bufG    = ws; ws += (size_t)BATCH * 5 * NPTS;       // xt
  float* partial = ws; ws += (size_t)BATCH * 1024 * NBLK64;
  float* gvec    = ws; ws += (size_t)BATCH * 1024;
  float* fc1     = ws; ws += (size_t)BATCH * 512;
  float* fc2     = ws; ws += (size_t)BATCH * 256;
  float* rpart   = ws; ws += (size_t)BATCH * 16 * 128 * NBLK64;
  float* pooled  = ws; ws += (size_t)BATCH * 16 * 128;

  dim3 blk256(256), blk32(32);

  // ---- STN(x, k=3) ----
  conv5_bn_relu_kernel<<<dim3(NB, 64, BATCH), blk256, 0, stream>>>(
      x, F(2), F(3), F(4), F(5), F(6), F(7), bufA, 64);
  convbn_wmma_kernel<<<dim3(NPTS / 16, 8, BATCH), blk32, 0, stream>>>(
      bufA, (size_t)64 * NPTS, F(8), F(9), F(10), F(11), F(12), F(13), bufB, 64, 128, 1);
  convbn_max_wmma_kernel<<<dim3(NBLK64, 8, BATCH), blk256, 0, stream>>>(
      bufB, F(14), F(15), F(16), F(17), F(18), F(19), partial, 1024, 1);
  reduce_max_kernel<<<dim3((BATCH * 1024 + 255) / 256), blk256, 0, stream>>>(
      partial, gvec, BATCH * 1024, NBLK64);
  fcbn_kernel<<<dim3((BATCH * 512 + 255) / 256), blk256, 0, stream>>>(
      gvec, F(20), F(21), F(22), F(23), F(24), F(25), fc1, 1024, 512);
  fcbn_kernel<<<dim3((BATCH * 256 + 255) / 256), blk256, 0, stream>>>(
      fc1, F(26), F(27), F(28), F(29), F(30), F(31), fc2, 512, 256);
  fceye_kernel<<<dim3(1), blk256, 0, stream>>>(fc2, F(32), F(33), trans, 256, 9, 3);

  // ---- apply transform, conv1 ----
  apply_trans_kernel<<<dim3(NB, BATCH), blk256, 0, stream>>>(x, trans, bufG);
  conv5_bn_relu_kernel<<<dim3(NB, 64, BATCH), blk256, 0, stream>>>(
      bufG, F(34), F(35), F(36), F(37), F(38), F(39), bufH, 64);

  // ---- FSTN(h, k=64) ----
  convbn_wmma_kernel<<<dim3(NPTS / 16, 4, BATCH), blk32, 0, stream>>>(
      bufH, (size_t)64 * NPTS, F(40), F(41), F(42), F(43), F(44), F(45), bufA, 64, 64, 1);
  convbn_wmma_kernel<<<dim3(NPTS / 16, 8, BATCH), blk32, 0, stream>>>(
      bufA, (size_t)64 * NPTS, F(46), F(47), F(48), F(49), F(50), F(51), bufB, 64, 128, 1);
  convbn_max_wmma_kernel<<<dim3(NBLK64, 8, BATCH), blk256, 0, stream>>>(
      bufB, F(52), F(53), F(54), F(55), F(56), F(57), partial, 1024, 1);
  reduce_max_kernel<<<dim3((BATCH * 1024 + 255) / 256), blk256, 0, stream>>>(
      partial, gvec, BATCH * 1024, NBLK64);
  fcbn_kernel<<<dim3((BATCH * 512 + 255) / 256), blk256, 0, stream>>>(
      gvec, F(58), F(59), F(60), F(61), F(62), F(63), fc1, 1024, 512);
  fcbn_kernel<<<dim3((BATCH * 256 + 255) / 256), blk256, 0, stream>>>(
      fc1, F(64), F(65), F(66), F(67), F(68), F(69), fc2, 512, 256);
  fceye_kernel<<<dim3((BATCH * 4096 + 255) / 256), blk256, 0, stream>>>(
      fc2, F(70), F(71), tf, 256, 4096, 64);

  // ---- h' = TF^T x h, written directly into output rows 1152:1216 ----
  feattrans_wmma_kernel<<<dim3(NPTS / 16, 4, BATCH), blk32, 0, stream>>>(bufH, tf, out);

  // ---- global pooling branch (reads h' from the output tensor) ----
  convbn_wmma_kernel<<<dim3(NPTS / 16, 8, BATCH), blk32, 0, stream>>>(
      out + (size_t)1152 * NPTS, (size_t)OUT_CH * NPTS,
      F(72), F(73), F(74), F(75), F(76), F(77), bufB, 64, 128, 1);
  convbn_max_wmma_kernel<<<dim3(NBLK64, 8, BATCH), blk256, 0, stream>>>(
      bufB, F(78), F(79), F(80), F(81), F(82), F(83), partial, 1024, 0 /*no relu*/);
  reduce_max_kernel<<<dim3((BATCH * 1024 + 255) / 256), blk256, 0, stream>>>(
      partial, gvec, BATCH * 1024, NBLK64);
  broadcast_kernel<<<dim3(NB, 1024, BATCH), blk256, 0, stream>>>(gvec, out);

  // ---- ring pooling branch ----
  ring_wmma_kernel<<<dim3(NBLK64, 16, BATCH), blk256, 0, stream>>>(out, F(84), ring, rpart);
  ring_reduce_bn_kernel<<<dim3((BATCH * 16 * 128 + 255) / 256), blk256, 0, stream>>>(
      rpart, F(85), F(86), F(87), F(88), F(89), pooled);
  ring_scatter_kernel<<<dim3(NB, 128, BATCH), blk256, 0, stream>>>(pooled, ring, out);
}